// SSIM_43301860278374
// MI455X (gfx1250) — compile-verified
//
#include <hip/hip_runtime.h>

typedef __attribute__((ext_vector_type(2))) float v2f;
typedef __attribute__((ext_vector_type(4))) float v4f;
typedef __attribute__((ext_vector_type(8))) float v8f;
typedef __attribute__((ext_vector_type(4))) int   v4i;

#define TILE 16
#define WPB 4                 // waves per block (wave32)
#define STRIP_W (TILE * WPB)  // 64 output cols per block
#define PATCH_H 28            // 16 + 10 halo, padded to K=28 for WMMA
#define PATCH_W 84            // 64 + 20 col halo (8 left, 12 right) -> 16B-aligned chunks
#define CHALO 8               // left column halo (16B aligned vs image)
#define CSHIFT 3              // patch col = output col + (CHALO - 5)
#define CHUNKS 21             // PATCH_W / 4 async chunks per row
#define TV_W 36               // TV staging stride (row-pair offset 2*36=72 -> disjoint banks)
#define IMG 512
#define PLANES 48             // 16 * 3
#define NPIX 12582912.0f      // 16*3*512*512

#if __has_builtin(__builtin_amdgcn_global_load_async_to_lds_b128)
#define USE_ASYNC_LDS 1
#else
#define USE_ASYNC_LDS 0
#endif

__global__ __launch_bounds__(128) void ssim_main(const float* __restrict__ A,
                                                 const float* __restrict__ B,
                                                 float* __restrict__ partial) {
  __shared__ alignas(16) float Ap[PATCH_H][PATCH_W];
  __shared__ alignas(16) float Bp[PATCH_H][PATCH_W];
  __shared__ float tv[WPB][16][TV_W];
  __shared__ float red[128];

  const int tid  = threadIdx.x;
  const int w    = tid >> 5;        // wave id 0..3
  const int lane = tid & 31;
  const int h    = lane >> 4;       // lane half (WMMA layout)
  const int li   = lane & 15;       // M (A-matrix) / N (B-matrix) index

  const size_t pbase = (size_t)blockIdx.z * IMG * IMG;
  const int R0 = blockIdx.y * TILE;
  const int C0 = blockIdx.x * STRIP_W;

  // ---- Patch fill: async global->LDS (ASYNCcnt path), zero-fill OOB chunks ----
#if USE_ASYNC_LDS
  for (int idx = tid; idx < PATCH_H * CHUNKS; idx += 128) {
    int pr = idx / CHUNKS, q = idx - pr * CHUNKS;
    int pc = 4 * q;
    int gr = R0 + pr - 5, gc = C0 + pc - CHALO;
    bool in = (gr >= 0) & (gr < IMG) & (gc >= 0) & (gc < IMG);  // gc%4==0, 512%4==0
    if (in) {
      size_t off = pbase + (size_t)gr * IMG + gc;
      __builtin_amdgcn_global_load_async_to_lds_b128((v4i*)(A + off),
                                                     (v4i*)&Ap[pr][pc], 0, 0);
      __builtin_amdgcn_global_load_async_to_lds_b128((v4i*)(B + off),
                                                     (v4i*)&Bp[pr][pc], 0, 0);
    } else {  // disjoint addresses from async writes -> no cross-counter hazard
      *(v4f*)&Ap[pr][pc] = (v4f){};
      *(v4f*)&Bp[pr][pc] = (v4f){};
    }
  }
#if __has_builtin(__builtin_amdgcn_s_wait_asynccnt)
  __builtin_amdgcn_s_wait_asynccnt(0);
#else
  asm volatile("s_wait_asynccnt 0x0" ::: "memory");
#endif
#else   // fallback: synchronous fill through VGPRs
  for (int idx = tid; idx < PATCH_H * PATCH_W; idx += 128) {
    int pr = idx / PATCH_W, pc = idx - pr * PATCH_W;
    int gr = R0 + pr - 5, gc = C0 + pc - CHALO;
    bool in = (gr >= 0) & (gr < IMG) & (gc >= 0) & (gc < IMG);
    float av = 0.f, bv = 0.f;
    if (in) {
      size_t off = pbase + (size_t)gr * IMG + gc;
      av = A[off];
      bv = B[off];
    }
    Ap[pr][pc] = av;
    Bp[pr][pc] = bv;
  }
#endif

  // Gaussian taps (sigma=1.5, K=11), normalized; constant-folds to literals.
  float gw[11];
  {
    float s = 0.f;
#pragma unroll
    for (int i = 0; i < 11; ++i) {
      float d = (float)(i - 5);
      gw[i] = expf(-d * d / 4.5f);
      s += gw[i];
    }
#pragma unroll
    for (int i = 0; i < 11; ++i) gw[i] /= s;
  }

  // A-fragments of the banded Gaussian G[m][j] = g[j-m], j = 4*kc + v + 2*h.
  // Same fragments serve both conv stages (Gv and Gh^T are the same band);
  // the band's zeros also kill the K=26,27 padding lanes automatically.
  float af[7][2];
#pragma unroll
  for (int kc = 0; kc < 7; ++kc) {
#pragma unroll
    for (int v = 0; v < 2; ++v) {
      int j = 4 * kc + v + 2 * h;
      int d = j - li;
      float val = 0.f;
#pragma unroll
      for (int t = 0; t < 11; ++t) val = (d == t) ? gw[t] : val;
      af[kc][v] = val;
    }
  }

  __syncthreads();

  // ---- Stage V: TV = Gv * P for 5 fields, two N-tiles (cols 0-15, 16-31) ----
  v8f acc[5][2];
#pragma unroll
  for (int f = 0; f < 5; ++f) {
    acc[f][0] = (v8f){};
    acc[f][1] = (v8f){};
  }
  const int tc = w * TILE;  // this wave's tile within the strip

#pragma unroll
  for (int kc = 0; kc < 7; ++kc) {
    v2f afr;
    afr.x = af[kc][0];
    afr.y = af[kc][1];
    int r0 = 4 * kc + 2 * h;  // B-matrix rows for v=0,1: r0, r0+1 (<= 27 < PATCH_H)
#pragma unroll
    for (int nt = 0; nt < 2; ++nt) {
      int pc = tc + 16 * nt + li + CSHIFT;  // <= 82 < PATCH_W
      float a0 = Ap[r0][pc], a1 = Ap[r0 + 1][pc];
      float b0 = Bp[r0][pc], b1 = Bp[r0 + 1][pc];
      v2f fr[5];
      fr[0] = (v2f){a0, a1};
      fr[1] = (v2f){b0, b1};
      fr[2] = (v2f){a0 * a0, a1 * a1};
      fr[3] = (v2f){b0 * b0, b1 * b1};
      fr[4] = (v2f){a0 * b0, a1 * b1};
#pragma unroll
      for (int f = 0; f < 5; ++f) {
        acc[f][nt] = __builtin_amdgcn_wmma_f32_16x16x4_f32(
            false, afr, false, fr[f], (short)0, acc[f][nt], false, false);
      }
    }
  }

  // ---- Stage H: OUT^T = Gh^T * TV^T per field, via per-wave LDS staging ----
  v8f res[5];
#pragma unroll
  for (int f = 0; f < 5; ++f) {
    __syncthreads();
#pragma unroll
    for (int i = 0; i < 8; ++i) {  // D-layout: row = i + 8*h, col = li (+16)
      tv[w][i + 8 * h][li]      = acc[f][0][i];
      tv[w][i + 8 * h][16 + li] = acc[f][1][i];
    }
    __syncthreads();
    v8f o = (v8f){};
#pragma unroll
    for (int kc = 0; kc < 7; ++kc) {
      v2f afr;
      afr.x = af[kc][0];
      afr.y = af[kc][1];
      int c0 = 4 * kc + 2 * h;  // TV column = K index (<= 27; band zeros K>=26)
      v2f bf;
      bf.x = tv[w][li][c0];
      bf.y = tv[w][li][c0 + 1];
      o = __builtin_amdgcn_wmma_f32_16x16x4_f32(false, afr, false, bf,
                                                (short)0, o, false, false);
    }
    res[f] = o;  // transposed tile; irrelevant for the global mean
  }

  // ---- Elementwise SSIM + per-lane partial sum ----
  const float C1v = 1.0e-4f, C2v = 9.0e-4f;
  float sum = 0.f;
#pragma unroll
  for (int i = 0; i < 8; ++i) {
    float mu1 = res[0][i], mu2 = res[1][i];
    float x2 = res[2][i], y2 = res[3][i], xy = res[4][i];
    float m11 = mu1 * mu1, m22 = mu2 * mu2, m12 = mu1 * mu2;
    float s1 = x2 - m11, s2 = y2 - m22, s12 = xy - m12;
    float num = (2.f * m12 + C1v) * (2.f * s12 + C2v);
    float den = (m11 + m22 + C1v) * (s1 + s2 + C2v);
    sum += num / den;
  }

  // Block tree reduction -> deterministic per-block partial.
  red[tid] = sum;
  __syncthreads();
#pragma unroll
  for (int s = 64; s > 0; s >>= 1) {
    if (tid < s) red[tid] += red[tid + s];
    __syncthreads();
  }
  if (tid == 0) {
    int bid = blockIdx.x + gridDim.x * (blockIdx.y + gridDim.y * blockIdx.z);
    partial[bid] = red[0];
  }
}

__global__ __launch_bounds__(256) void ssim_reduce(const float* __restrict__ partial,
                                                   float* __restrict__ out, int n) {
  __shared__ float red[256];
  float s = 0.f;
  for (int i = threadIdx.x; i < n; i += 256) s += partial[i];
  red[threadIdx.x] = s;
  __syncthreads();
#pragma unroll
  for (int t = 128; t > 0; t >>= 1) {
    if (threadIdx.x < t) red[threadIdx.x] += red[threadIdx.x + t];
    __syncthreads();
  }
  if (threadIdx.x == 0) out[0] = red[0] / NPIX;
}

extern "C" void kernel_launch(void* const* d_in, const int* in_sizes, int n_in,
                              void* d_out, int out_size, void* d_ws, size_t ws_size,
                              hipStream_t stream) {
  const float* A = (const float*)d_in[0];
  const float* B = (const float*)d_in[1];
  float* out = (float*)d_out;
  float* partial = (float*)d_ws;  // 8*32*48 = 12288 floats = 48 KB

  dim3 grid(IMG / STRIP_W, IMG / TILE, PLANES);  // (8, 32, 48)
  ssim_main<<<grid, 128, 0, stream>>>(A, B, partial);
  ssim_reduce<<<1, 256, 0, stream>>>(partial, out, 8 * 32 * 48);
}